// ObjectRepresentationSNN_8117488189518
// MI455X (gfx1250) — compile-verified
//
#include <hip/hip_runtime.h>
#include <hip/hip_bf16.h>

#define B_ 16
#define N_ 1024
#define D_ 64
#define STEPS_ 12
#define NUM_CLASSES_ 10

typedef float v2f __attribute__((ext_vector_type(2)));
typedef float v8f __attribute__((ext_vector_type(8)));

// -------- encode: encoded[b,n,d] = x[b,n,:3] @ enc_w + enc_b --------
__global__ __launch_bounds__(256) void k_encode(const float* __restrict__ x,
                                                const float* __restrict__ enc_w,
                                                const float* __restrict__ enc_b,
                                                float* __restrict__ encoded) {
    int i = blockIdx.x * 256 + threadIdx.x;          // over B*N*D
    if (i >= B_ * N_ * D_) return;
    int d = i & (D_ - 1);
    int bn = i >> 6;
    const float* xp = x + bn * 3;
    float v = enc_b[d];
    v = fmaf(xp[0], enc_w[0 * D_ + d], v);
    v = fmaf(xp[1], enc_w[1 * D_ + d], v);
    v = fmaf(xp[2], enc_w[2 * D_ + d], v);
    encoded[i] = v;
}

// -------- WMMA f32 GEMM: tmp = encoded + theta @ gr_w + gr_b --------
// theta viewed as [16384, 64]; one wave computes a full 16x64 row-block:
// A (16x4 f32) loaded once per K-slice, reused across 4 independent
// accumulator chains (one per 16-col tile) -> 64 WMMAs per wave, 4-way ILP.
// A layout (ISA 7.12.2): lane l -> M = l%16, K = 4*kk + 2*(l/16) + v
// C/D layout:            lane l, vgpr v -> M = v + 8*(l/16), N = l%16
__global__ __launch_bounds__(32) void k_gemm(const float* __restrict__ theta,
                                             const float* __restrict__ gr_w,
                                             const float* __restrict__ gr_b,
                                             const float* __restrict__ encoded,
                                             float* __restrict__ tmp) {
    const int l = threadIdx.x;
    const int row_base = blockIdx.x * 16;            // M tile over 16384 rows
    const int m = l & 15;
    const int half = l >> 4;
    v8f acc0 = {}, acc1 = {}, acc2 = {}, acc3 = {};
    const float* arow = theta + (size_t)(row_base + m) * D_;
#pragma unroll
    for (int kk = 0; kk < 16; ++kk) {
        const int kb = kk * 4 + half * 2;
        v2f a;
        a.x = arow[kb];
        a.y = arow[kb + 1];
        const float* brow0 = gr_w + (size_t)kb * D_ + m;        // K = kb,   N = m + 16*ctile
        const float* brow1 = gr_w + (size_t)(kb + 1) * D_ + m;  // K = kb+1
        v2f b0, b1, b2, b3;
        b0.x = brow0[0];  b0.y = brow1[0];
        b1.x = brow0[16]; b1.y = brow1[16];
        b2.x = brow0[32]; b2.y = brow1[32];
        b3.x = brow0[48]; b3.y = brow1[48];
        acc0 = __builtin_amdgcn_wmma_f32_16x16x4_f32(false, a, false, b0, (short)0, acc0, false, false);
        acc1 = __builtin_amdgcn_wmma_f32_16x16x4_f32(false, a, false, b1, (short)0, acc1, false, false);
        acc2 = __builtin_amdgcn_wmma_f32_16x16x4_f32(false, a, false, b2, (short)0, acc2, false, false);
        acc3 = __builtin_amdgcn_wmma_f32_16x16x4_f32(false, a, false, b3, (short)0, acc3, false, false);
    }
#pragma unroll
    for (int v = 0; v < 8; ++v) {
        const size_t row = (size_t)(row_base + v + half * 8) * D_;
        tmp[row + m]      = acc0[v] + encoded[row + m]      + gr_b[m];
        tmp[row + 16 + m] = acc1[v] + encoded[row + 16 + m] + gr_b[16 + m];
        tmp[row + 32 + m] = acc2[v] + encoded[row + 32 + m] + gr_b[32 + m];
        tmp[row + 48 + m] = acc3[v] + encoded[row + 48 + m] + gr_b[48 + m];
    }
}

// -------- per-batch group sums of sin/cos(theta) keyed by prev spike --------
// gsum[b][{S0,C0,S1,C1}][d]
__global__ __launch_bounds__(256) void k_groupsum(const float* __restrict__ theta,
                                                  const float* __restrict__ spk,
                                                  float* __restrict__ gsum) {
    __shared__ float lds[4][4][D_];                  // chunk, {S0,C0,S1,C1}, d
    const int b = blockIdx.x;
    const int tid = threadIdx.x;
    const int d = tid & (D_ - 1);
    const int chunk = tid >> 6;
    float S0 = 0.f, C0 = 0.f, S1 = 0.f, C1 = 0.f;
    const float* th = theta + (size_t)b * N_ * D_;
    const float* sp = spk + (size_t)b * N_;
    const int n0 = chunk * (N_ / 4);
    for (int n = n0; n < n0 + N_ / 4; ++n) {
        float t = th[(size_t)n * D_ + d];
        float s = __sinf(t), c = __cosf(t);
        if (sp[n] > 0.5f) { S1 += s; C1 += c; } else { S0 += s; C0 += c; }
    }
    lds[chunk][0][d] = S0; lds[chunk][1][d] = C0;
    lds[chunk][2][d] = S1; lds[chunk][3][d] = C1;
    __syncthreads();
    if (tid < D_) {
        float a0 = 0.f, a1 = 0.f, a2 = 0.f, a3 = 0.f;
#pragma unroll
        for (int k = 0; k < 4; ++k) {
            a0 += lds[k][0][d]; a1 += lds[k][1][d];
            a2 += lds[k][2][d]; a3 += lds[k][3][d];
        }
        float* g = gsum + (size_t)b * 4 * D_;
        g[0 * D_ + d] = a0; g[1 * D_ + d] = a1;
        g[2 * D_ + d] = a2; g[3 * D_ + d] = a3;
    }
}

// -------- fused per-node update: one wave32 per (b,n) --------
__global__ __launch_bounds__(256) void k_update(const float* __restrict__ tmp,
                                                const float* __restrict__ gamma_gain,
                                                const float* __restrict__ in_w,
                                                const float* __restrict__ in_b_p,
                                                const float* __restrict__ gsum,
                                                float* __restrict__ theta,
                                                float* __restrict__ hist0,
                                                float* __restrict__ hist1,
                                                float* __restrict__ mem,
                                                float* __restrict__ spk,
                                                float* __restrict__ rate_sum,
                                                int t) {
    const int wave = blockIdx.x * (blockDim.x >> 5) + (threadIdx.x >> 5);
    const int lane = threadIdx.x & 31;
    const int b = wave >> 10;
    const int n = wave & (N_ - 1);
    const size_t base = (size_t)wave * D_;

    float* hist = (t & 1) ? hist1 : hist0;           // holds theta_end(t-2); overwritten after read
    const float spk_prev = spk[wave];
    const float* gs = gsum + (size_t)b * 4 * D_ + (spk_prev > 0.5f ? 2 * D_ : 0);
    const float kcoup = (t == 0) ? 0.f : (0.5f / (float)N_);

    float curr_part = 0.f;
#pragma unroll
    for (int r = 0; r < 2; ++r) {
        const int d = lane + r * 32;
        const float th  = theta[base + d];
        const float pre = tmp[base + d];
        const float gamma = tanhf(fabsf(pre) * gamma_gain[(size_t)n * D_ + d]);
        const float s = __sinf(th), c = __cosf(th);
        const float coup = c * gs[d] - s * gs[D_ + d];     // c*S_g - s*C_g (diag cancels)
        const float th_new = th + 0.1f * (gamma + kcoup * coup);
        const float th_del = (t >= 2) ? hist[base + d] : th;
        const float gate = 0.5f * (1.f + __sinf(th_del));
        curr_part = fmaf(gate * gamma, in_w[d], curr_part);
        theta[base + d] = th_new;
        hist[base + d] = th_new;                           // push ring slot t%2
    }
#pragma unroll
    for (int off = 16; off > 0; off >>= 1)
        curr_part += __shfl_down(curr_part, off, 32);
    if (lane == 0) {
        const float current = curr_part + in_b_p[0];
        const float m_new = 0.9f * mem[wave] + current - spk_prev; // RESET_SCALE*THRESH = 1
        const float s_new = (m_new > 1.0f) ? 1.f : 0.f;
        mem[wave] = m_new;
        spk[wave] = s_new;
        rate_sum[wave] += s_new;
    }
}

// -------- classifier: logits = (rate_sum/12) @ cls_w + cls_b --------
__global__ __launch_bounds__(256) void k_classify(const float* __restrict__ rate_sum,
                                                  const float* __restrict__ cls_w,
                                                  const float* __restrict__ cls_b,
                                                  float* __restrict__ out) {
    __shared__ float red[256];
    const int b = blockIdx.x / NUM_CLASSES_;
    const int k = blockIdx.x % NUM_CLASSES_;
    const int tid = threadIdx.x;
    float acc = 0.f;
    for (int n = tid; n < N_; n += 256)
        acc = fmaf(rate_sum[(size_t)b * N_ + n] * (1.f / (float)STEPS_),
                   cls_w[(size_t)n * NUM_CLASSES_ + k], acc);
    red[tid] = acc;
    __syncthreads();
    for (int s2 = 128; s2 > 0; s2 >>= 1) {
        if (tid < s2) red[tid] += red[tid + s2];
        __syncthreads();
    }
    if (tid == 0) out[b * NUM_CLASSES_ + k] = red[0] + cls_b[k];
}

extern "C" void kernel_launch(void* const* d_in, const int* in_sizes, int n_in,
                              void* d_out, int out_size, void* d_ws, size_t ws_size,
                              hipStream_t stream) {
    const float* x          = (const float*)d_in[0];
    const float* enc_w      = (const float*)d_in[1];
    const float* enc_b      = (const float*)d_in[2];
    const float* gr_w       = (const float*)d_in[3];
    const float* gr_b       = (const float*)d_in[4];
    const float* gamma_gain = (const float*)d_in[5];
    const float* in_w       = (const float*)d_in[6];
    const float* in_b       = (const float*)d_in[7];
    const float* cls_w      = (const float*)d_in[8];
    const float* cls_b      = (const float*)d_in[9];
    float* out = (float*)d_out;

    const size_t BND = (size_t)B_ * N_ * D_;   // 1,048,576
    const size_t BN  = (size_t)B_ * N_;        // 16,384
    float* ws       = (float*)d_ws;
    float* encoded  = ws;               ws += BND;
    float* tmp      = ws;               ws += BND;
    float* theta    = ws;               ws += BND;
    float* hist0    = ws;               ws += BND;
    float* hist1    = ws;               ws += BND;
    float* mem      = ws;               ws += BN;
    float* spk      = ws;               ws += BN;
    float* rate_sum = ws;               ws += BN;
    float* gsum     = ws;               ws += (size_t)B_ * 4 * D_;

    hipMemsetAsync(theta,    0, BND * sizeof(float), stream);
    hipMemsetAsync(mem,      0, BN * sizeof(float),  stream);
    hipMemsetAsync(spk,      0, BN * sizeof(float),  stream);
    hipMemsetAsync(rate_sum, 0, BN * sizeof(float),  stream);

    k_encode<<<(int)((BND + 255) / 256), 256, 0, stream>>>(x, enc_w, enc_b, encoded);

    for (int t = 0; t < STEPS_; ++t) {
        k_gemm<<<(int)(BN / 16), 32, 0, stream>>>(theta, gr_w, gr_b, encoded, tmp);
        k_groupsum<<<B_, 256, 0, stream>>>(theta, spk, gsum);
        k_update<<<(int)(BN / 8), 256, 0, stream>>>(tmp, gamma_gain, in_w, in_b, gsum,
                                                    theta, hist0, hist1, mem, spk, rate_sum, t);
    }

    k_classify<<<B_ * NUM_CLASSES_, 256, 0, stream>>>(rate_sum, cls_w, cls_b, out);
}